// GraphConv_51041391345939
// MI455X (gfx1250) — compile-verified
//
#include <hip/hip_runtime.h>
#include <hip/hip_bf16.h>
#include <cstdint>

#define EMB 64

// ---------------------------------------------------------------------------
// Edge-parallel COO SpMM: y[r,:] += v * x[c,:].
// 16 lanes per edge, each lane owns one 16B (float4) chunk of the 64-float row.
// - All index/offset math in 32-bit unsigned (total threads ~52M < 2^32) so
//   index loads use SADDR + zext-voffset addressing, no 64-bit VALU chains.
// - Index streams (rows/cols/vals) are single-use: nontemporal loads so the
//   ~234 MB of streamed indices never evict the L2-resident embeddings.
// - The x[c] gather is staged through LDS via the gfx1250 async-load path
//   (global_load_async_to_lds_b128 -> s_wait_asynccnt -> ds read of own slot).
// - Scatter uses no-return global_atomic_add_f32 (resolved at L2 atomic units).
// ---------------------------------------------------------------------------
__global__ void spmm_edge_kernel(const int* __restrict__ rows,
                                 const int* __restrict__ cols,
                                 const float* __restrict__ vals,
                                 const float* __restrict__ x,
                                 float* __restrict__ y,
                                 unsigned nnz) {
    __shared__ float4 stage[256];   // 256 threads * 16B = 4KB
    const unsigned tid = threadIdx.x;
    const unsigned gt = blockIdx.x * 256u + tid;  // fits in 32 bits
    const unsigned e     = gt >> 4;               // edge index
    const unsigned chunk = gt & 15u;              // which float4 of the row
    const bool active = (e < nnz);

    unsigned col = 0, row = 0;
    float v = 0.0f;
    if (active) {
        // 16 consecutive lanes hit the same 4B address -> single request.
        // Nontemporal: streamed once, keep out of the hot L2 set.
        col = (unsigned)__builtin_nontemporal_load(cols + e);
        row = (unsigned)__builtin_nontemporal_load(rows + e);
        v   = __builtin_nontemporal_load(vals + e);
    }

    // Async gather of my 16B chunk of x[col] into my private LDS slot.
    // Inactive tail lanes use a clamped (valid) address and discard the data.
    const float* gsrc = x + (col * EMB + chunk * 4u);  // 32-bit offset, zext
    // Low 32 bits of the generic pointer to LDS == byte offset within LDS.
    unsigned lds_off = (unsigned)(uintptr_t)(&stage[tid]);
    asm volatile("global_load_async_to_lds_b128 %0, %1, off"
                 :: "v"(lds_off), "v"(gsrc)
                 : "memory");
    asm volatile("s_wait_asynccnt 0x0" ::: "memory");

    const float4 r = stage[tid];  // ds_load_b128 of my own slot; no barrier

    if (active) {
        float* dst = y + (row * EMB + chunk * 4u);     // 32-bit offset, zext
        unsafeAtomicAdd(dst + 0, r.x * v);
        unsafeAtomicAdd(dst + 1, r.y * v);
        unsafeAtomicAdd(dst + 2, r.z * v);
        unsafeAtomicAdd(dst + 3, r.w * v);
    }
}

// scatter target must start from 0 each layer
__global__ void zero4_kernel(float4* __restrict__ p, int n4) {
    int i = blockIdx.x * blockDim.x + threadIdx.x;
    if (i < n4) p[i] = make_float4(0.f, 0.f, 0.f, 0.f);
}

// x_cur = ego ; acc = ego   (acc lives directly in d_out)
__global__ void init2_kernel(const float4* __restrict__ src,
                             float4* __restrict__ a,
                             float4* __restrict__ b, int n4) {
    int i = blockIdx.x * blockDim.x + threadIdx.x;
    if (i < n4) { float4 v = src[i]; a[i] = v; b[i] = v; }
}

// acc = (acc + x) * s   (s = 1 for inner layers, 0.25 fused on the last layer)
__global__ void addscale_kernel(float4* __restrict__ acc,
                                const float4* __restrict__ x,
                                float s, int n4) {
    int i = blockIdx.x * blockDim.x + threadIdx.x;
    if (i < n4) {
        float4 a = acc[i];
        float4 b = x[i];
        a.x = (a.x + b.x) * s;
        a.y = (a.y + b.y) * s;
        a.z = (a.z + b.z) * s;
        a.w = (a.w + b.w) * s;
        acc[i] = a;
    }
}

extern "C" void kernel_launch(void* const* d_in, const int* in_sizes, int n_in,
                              void* d_out, int out_size, void* d_ws, size_t ws_size,
                              hipStream_t stream) {
    const float* ego   = (const float*)d_in[0];
    const int*   rows1 = (const int*)  d_in[1];
    const int*   cols1 = (const int*)  d_in[2];
    const float* vals1 = (const float*)d_in[3];
    const int*   rows2 = (const int*)  d_in[4];
    const int*   cols2 = (const int*)  d_in[5];
    const float* vals2 = (const float*)d_in[6];
    // d_in[7] = n_users: unused, output layout is contiguous (v1 full, v2 full)

    const int LAYERS = 3;
    const int N   = in_sizes[0] / EMB;          // 100000 nodes
    const int nnz = in_sizes[1] / LAYERS;       // edges per layer
    const int n4  = N * EMB / 4;                // float4 elements per matrix

    float* xa = (float*)d_ws;                   // ping
    float* xb = xa + (size_t)N * EMB;           // pong  (2 x 25.6 MB in d_ws)
    float* out = (float*)d_out;

    const dim3 blk(256);
    const int vgrid = (n4 + 255) / 256;
    const long long ethreads = (long long)nnz * 16;
    const int egrid = (int)((ethreads + 255) / 256);

    for (int chain = 0; chain < 2; ++chain) {
        const int*   R = chain ? rows2 : rows1;
        const int*   C = chain ? cols2 : cols1;
        const float* V = chain ? vals2 : vals1;
        float* acc = out + (size_t)chain * N * EMB;

        init2_kernel<<<vgrid, blk, 0, stream>>>((const float4*)ego,
                                                (float4*)xa, (float4*)acc, n4);
        float* xc = xa;
        float* xn = xb;
        for (int l = 0; l < LAYERS; ++l) {
            zero4_kernel<<<vgrid, blk, 0, stream>>>((float4*)xn, n4);
            spmm_edge_kernel<<<egrid, blk, 0, stream>>>(R + (size_t)l * nnz,
                                                        C + (size_t)l * nnz,
                                                        V + (size_t)l * nnz,
                                                        xc, xn, (unsigned)nnz);
            const float s = (l == LAYERS - 1) ? 0.25f : 1.0f;
            addscale_kernel<<<vgrid, blk, 0, stream>>>((float4*)acc,
                                                       (const float4*)xn, s, n4);
            float* t = xc; xc = xn; xn = t;
        }
    }
}